// Masker_inputfree_23888608101163
// MI455X (gfx1250) — compile-verified
//
#include <hip/hip_runtime.h>
#include <hip/hip_bf16.h>
#include <cstddef>
#include <cstdint>

// ---------------------------------------------------------------------------
// Masker (input-free) forward on MI455X / gfx1250, wave32.
//
//   y1   = relu(bn(x @ W1 + b1))          x:(1,2048)  W1:(2048,8192)
//   y2   = relu(bn(y1 @ W2 + b2))                     W2:(8192,8192)
//   mask = bn(y2 @ W3 + b3, g=1, beta=0)              W3:(8192,2048)
//   z[c] = max_k st[k][c],  st = straight-through hard gumbel-softmax
//
// 384 MB of single-use f32 weights -> HBM-bandwidth bound (~16.5 us floor at
// 23.3 TB/s). Weight streaming is done by the Tensor Data Mover: strided 2D
// tiles (32 cols x 64 rows, full 128B lines per row) DMA'd into LDS with
// double buffering, while the wave runs V_WMMA_F32_16X16X4_F32 out of LDS.
// ---------------------------------------------------------------------------

typedef float v2f  __attribute__((ext_vector_type(2)));
typedef float v8f  __attribute__((ext_vector_type(8)));
typedef unsigned int u32x4 __attribute__((ext_vector_type(4)));
typedef int   i32x4 __attribute__((ext_vector_type(4)));
typedef int   i32x8 __attribute__((ext_vector_type(8)));

#define BN_EPS   1e-5f
#define INV_TAU  2.0f          // 1 / 0.5
#define NCLASS   2048
#define KSPLIT   16            // K-dimension split for memory-level parallelism
#define CHUNK_ROWS 64          // K-rows per TDM chunk
#define TILE_COLS  32          // N-columns per wave (two 16-wide WMMA tiles)
#define ROW_STRIDE 40          // LDS dwords/row: 32 data + 8 pad (bank decollide)

// ---------------------------------------------------------------------------
// TDM: issue TENSOR_LOAD_TO_LDS of a 2D tile (tile_w x tile_h f32 elements,
// row stride `stride_elems` in memory) into contiguous LDS at lds_addr.
// D# bitfields per CDNA5 ISA 8.3/8.4: group0 = {count|flags, lds_addr,
// global_addr[31:0], global_addr[56:32]|type=2}; group1 = {data_size=4B
// (+optional LDS padding), tensor dims, tile dims, dim0 stride}.
// pad: interval code 4 (=32 dwords = one row), amount code 7 (=8 dwords).
// This toolchain's builtin takes 6 args (g0, g1, g2, g3, g_extra, cpol);
// trailing groups are zero/disabled for 2D tiles.
// ---------------------------------------------------------------------------
__device__ __forceinline__ void tdm_load_2d(const void* gptr, uint32_t lds_addr,
                                            uint32_t tile_w, uint32_t tile_h,
                                            uint64_t stride_elems, int pad) {
  const uint64_t gaddr = (uint64_t)(uintptr_t)gptr;
  u32x4 g0;
  g0.x = 1u;                                              // count=1, user mode
  g0.y = lds_addr;                                        // lds_addr (bytes)
  g0.z = (uint32_t)gaddr;                                 // global_addr[31:0]
  g0.w = (uint32_t)((gaddr >> 32) & 0x01FFFFFFu)          // global_addr[56:32]
       | (2u << 30);                                      // type = 2 ("image")
  uint32_t d0 = 0x00020000u;                              // data_size = 4B
  if (pad) d0 |= (1u << 20) | (4u << 22) | (7u << 25);    // pad en|intv=32dw|amt=8dw
  i32x8 g1;
  g1[0] = (int)d0;                                        // wg_mask=0 (no cluster)
  g1[1] = (int)(tile_w << 16);                            // tensor_dim0[15:0] @ [63:48]
  g1[2] = (int)(tile_h << 16);                            // tdim0 hi=0; tensor_dim1 lo
  g1[3] = (int)(tile_w << 16);                            // tdim1 hi=0; tile_dim0
  g1[4] = (int)(tile_h & 0xFFFFu);                        // tile_dim1; tile_dim2=0
  g1[5] = (int)(uint32_t)stride_elems;                    // tensor_dim0_stride lo32
  g1[6] = (int)((stride_elems >> 32) & 0xFFFFu);          // stride hi16; dim1_stride=0
  g1[7] = 0;
  const i32x4 gz4 = {0, 0, 0, 0};                         // groups 2/3 unused (2D)
  const i32x8 gz8 = {0, 0, 0, 0, 0, 0, 0, 0};             // extra group unused
  __builtin_amdgcn_tensor_load_to_lds(g0, g1, gz4, gz4, gz8, 0);
}

// ---------------------------------------------------------------------------
// Partial matvec: one wave per (32-column tile, K-slice).
//   part[ks*N + n] = sum_{k in slice} x[k] * W[k][n]
//
// WMMA f32 16x16x4 operand placement (ISA 7.12.2, 32-bit layouts):
//   A (16x4): lane L -> M = L&15, K = 2*(L>=16) + vgpr   (row M=0 carries x)
//   B (4x16): lane L -> N = L&15, K = 2*(L>=16) + vgpr
//   D row M=0 -> VGPR 0, lanes 0..15
// B tiles come from LDS (TDM double-buffered). A's zero rows are produced by
// address selection: non-row0 lanes read a zeroed LDS slot, so every A build
// is one unconditional ds_load_b64 (no exec-mask branching).
// ---------------------------------------------------------------------------
__global__ void __launch_bounds__(32)
matvec_wmma_part(const float* __restrict__ x,
                 const float* __restrict__ W,   // K x N row-major
                 float* __restrict__ part,      // KSPLIT x N
                 int kslice, int N) {
  __shared__ float wb[2][CHUNK_ROWS * ROW_STRIDE];   // 2 x 10 KB weight buffers
  __shared__ float xs[512];                          // x slice (kslice <= 512)
  __shared__ float xz[2];                            // zero slot for A rows 1..15

  const int lane    = threadIdx.x;                   // full wave, EXEC all 1s
  const int n0      = blockIdx.x * TILE_COLS;
  const int ks      = blockIdx.y;
  const int kbeg    = ks * kslice;
  const int nchunks = kslice / CHUNK_ROWS;           // >= 2 for all layers

  const int  c15   = lane & 15;
  const int  khalf = (lane >> 4) << 1;               // 0 | 2 (K half select)
  const bool row0  = (c15 == 0);                     // lanes holding A's M=0

  if (lane < 2) xz[lane] = 0.0f;                     // same-wave LDS is in-order

  // Stage the x-slice (1D) and the first weight chunk.
  tdm_load_2d(x + kbeg, (uint32_t)(uintptr_t)xs, (uint32_t)kslice, 1,
              (uint64_t)kslice, 0);
  tdm_load_2d(W + (size_t)kbeg * (size_t)N + n0, (uint32_t)(uintptr_t)&wb[0][0],
              TILE_COLS, CHUNK_ROWS, (uint64_t)N, 1);

  v8f acc0 = {}, acc1 = {};

  for (int c = 0; c < nchunks; ++c) {
    const int buf = c & 1;
    if (c + 1 < nchunks) {
      // Prefetch next chunk into the other buffer, then wait for current.
      tdm_load_2d(W + (size_t)(kbeg + (c + 1) * CHUNK_ROWS) * (size_t)N + n0,
                  (uint32_t)(uintptr_t)&wb[buf ^ 1][0],
                  TILE_COLS, CHUNK_ROWS, (uint64_t)N, 1);
      __builtin_amdgcn_s_wait_tensorcnt(1);          // TDM completes in-order
    } else {
      __builtin_amdgcn_s_wait_tensorcnt(0);
    }

    const float* xsrc = &xs[c * CHUNK_ROWS];
    const float* wsrc = &wb[buf][0];

    // One k-step = 4 K-rows: A via single ds_load_b64 (address-selected),
    // B0/B1 via 2x ds_load_2addr pairs. 1-deep software pipeline so the LDS
    // latency of step r+4 hides under the two WMMAs of step r.
    auto load_step = [&](int r, v2f& a, v2f& b0, v2f& b1) {
      const int rr = r + khalf;
      const float* ap = row0 ? (xsrc + rr) : xz;     // zero slot for rows 1..15
      a.x = ap[0];                                   // K = rr
      a.y = ap[1];                                   // K = rr+1
      const float* wrow = wsrc + rr * ROW_STRIDE + c15;
      b0.x = wrow[0];
      b0.y = wrow[ROW_STRIDE];
      b1.x = wrow[16];
      b1.y = wrow[ROW_STRIDE + 16];
    };

    v2f a0, b00, b10, a1, b01, b11;
    load_step(0, a0, b00, b10);
#pragma unroll 5
    for (int r = 0; r < CHUNK_ROWS - 4; r += 4) {
      load_step(r + 4, a1, b01, b11);
      acc0 = __builtin_amdgcn_wmma_f32_16x16x4_f32(false, a0, false, b00,
                                                   (short)0, acc0, false, false);
      acc1 = __builtin_amdgcn_wmma_f32_16x16x4_f32(false, a0, false, b10,
                                                   (short)0, acc1, false, false);
      a0 = a1; b00 = b01; b10 = b11;
    }
    acc0 = __builtin_amdgcn_wmma_f32_16x16x4_f32(false, a0, false, b00,
                                                 (short)0, acc0, false, false);
    acc1 = __builtin_amdgcn_wmma_f32_16x16x4_f32(false, a0, false, b10,
                                                 (short)0, acc1, false, false);
  }

  // D row M=0 lives in acc[0], lanes 0..15.
  if (lane < 16) {
    part[(size_t)ks * (size_t)N + (size_t)(n0 + lane)]      = acc0[0];
    part[(size_t)ks * (size_t)N + (size_t)(n0 + 16 + lane)] = acc1[0];
  }
}

// ---------------------------------------------------------------------------
// Sum K-split partials deterministically, add bias, BN, optional ReLU.
// g/beta may be nullptr (=> gamma=1, beta=0, used for layer 3).
// ---------------------------------------------------------------------------
__global__ void finalize_bn(const float* __restrict__ part, int N, int ksplit,
                            const float* __restrict__ b,
                            const float* __restrict__ g,
                            const float* __restrict__ beta,
                            const float* __restrict__ m,
                            const float* __restrict__ v,
                            float* __restrict__ out, int do_relu) {
  const int n = blockIdx.x * blockDim.x + threadIdx.x;
  if (n >= N) return;
  float s = 0.0f;
  for (int i = 0; i < ksplit; ++i) s += part[(size_t)i * (size_t)N + n];
  s += b[n];
  float r = (s - m[n]) * rsqrtf(v[n] + BN_EPS);
  const float gg = g    ? g[n]    : 1.0f;
  const float bb = beta ? beta[n] : 0.0f;
  float val = r * gg + bb;
  if (do_relu) val = fmaxf(val, 0.0f);
  out[n] = val;
}

__global__ void zero_f32(float* __restrict__ p, int n) {
  const int i = blockIdx.x * blockDim.x + threadIdx.x;
  if (i < n) p[i] = 0.0f;
}

// ---------------------------------------------------------------------------
// One block per gumbel sample row (1024 rows, 2048 classes, 256 threads,
// 8 contiguous classes per thread so tie-breaks match jnp.argmax first-index).
//   l = (mask + (-log(-log(u)))) / tau
//   soft@argmax = 1/sum(exp(l - lmax)); st@argmax = (1+p)-p; st elsewhere = 0
//   z[argmax] = max(z[argmax], st) via bitwise atomicMax (values >= 0)
// ---------------------------------------------------------------------------
#define GT 256
__global__ void __launch_bounds__(GT)
gumbel_st_max(const float* __restrict__ mask,
              const float* __restrict__ u,
              float* __restrict__ z) {
  __shared__ float smax[GT];
  __shared__ int   sidx[GT];
  __shared__ float ssum[GT];

  const int row = blockIdx.x;
  const int t   = threadIdx.x;
  const float* urow = u + (size_t)row * NCLASS;
  const int base = t * (NCLASS / GT);

  float l[NCLASS / GT];
  float lm = -__builtin_inff();
  int   li = base;
#pragma unroll
  for (int i = 0; i < NCLASS / GT; ++i) {
    const int c = base + i;
    const float uu  = __builtin_nontemporal_load(urow + c);
    const float gmb = -__logf(-__logf(uu));
    const float lv  = (mask[c] + gmb) * INV_TAU;
    l[i] = lv;
    if (lv > lm) { lm = lv; li = c; }          // strict > : first occurrence
  }
  smax[t] = lm; sidx[t] = li;
  __syncthreads();

  for (int s = GT / 2; s > 0; s >>= 1) {
    if (t < s) {
      const float a = smax[t], b2 = smax[t + s];
      if (b2 > a || (b2 == a && sidx[t + s] < sidx[t])) {
        smax[t] = b2; sidx[t] = sidx[t + s];
      }
    }
    __syncthreads();
  }
  const float gmax = smax[0];
  const int   gidx = sidx[0];

  float sum = 0.0f;
#pragma unroll
  for (int i = 0; i < NCLASS / GT; ++i) sum += __expf(l[i] - gmax);
  ssum[t] = sum;
  __syncthreads();
  for (int s = GT / 2; s > 0; s >>= 1) {
    if (t < s) ssum[t] += ssum[t + s];
    __syncthreads();
  }

  if (t == 0) {
    const float p  = 1.0f / ssum[0];
    const float st = (1.0f + p) - p;           // match reference fp ordering
    atomicMax((unsigned int*)&z[gidx], __float_as_uint(st)); // st >= 0
  }
}

// ---------------------------------------------------------------------------
extern "C" void kernel_launch(void* const* d_in, const int* in_sizes, int n_in,
                              void* d_out, int out_size, void* d_ws, size_t ws_size,
                              hipStream_t stream) {
  (void)in_sizes; (void)n_in; (void)out_size; (void)ws_size;

  const float* fix  = (const float*)d_in[0];   // (1,2048)
  const float* W1   = (const float*)d_in[1];   // (2048,8192)
  const float* b1   = (const float*)d_in[2];
  const float* g1   = (const float*)d_in[3];
  const float* be1  = (const float*)d_in[4];
  const float* m1   = (const float*)d_in[5];
  const float* v1   = (const float*)d_in[6];
  const float* W2   = (const float*)d_in[7];   // (8192,8192)
  const float* b2   = (const float*)d_in[8];
  const float* g2   = (const float*)d_in[9];
  const float* be2  = (const float*)d_in[10];
  const float* m2   = (const float*)d_in[11];
  const float* v2   = (const float*)d_in[12];
  const float* W3   = (const float*)d_in[13];  // (8192,2048)
  const float* b3   = (const float*)d_in[14];
  const float* m3   = (const float*)d_in[15];
  const float* v3   = (const float*)d_in[16];
  const float* u    = (const float*)d_in[17];  // (1024,2048)

  float* z    = (float*)d_out;                 // (2048,)
  float* ws   = (float*)d_ws;
  float* part = ws;                            // KSPLIT * 8192 floats
  float* y1   = part + KSPLIT * 8192;          // 8192
  float* y2   = y1 + 8192;                     // 8192
  float* mk   = y2 + 8192;                     // 2048

  // Layer 1: (2048 -> 8192), kslice = 128 (2 TDM chunks/wave)
  matvec_wmma_part<<<dim3(8192 / TILE_COLS, KSPLIT), 32, 0, stream>>>(
      fix, W1, part, 2048 / KSPLIT, 8192);
  finalize_bn<<<(8192 + 255) / 256, 256, 0, stream>>>(
      part, 8192, KSPLIT, b1, g1, be1, m1, v1, y1, 1);

  // Layer 2: (8192 -> 8192), kslice = 512 (8 TDM chunks/wave)
  matvec_wmma_part<<<dim3(8192 / TILE_COLS, KSPLIT), 32, 0, stream>>>(
      y1, W2, part, 8192 / KSPLIT, 8192);
  finalize_bn<<<(8192 + 255) / 256, 256, 0, stream>>>(
      part, 8192, KSPLIT, b2, g2, be2, m2, v2, y2, 1);

  // Layer 3: (8192 -> 2048), gamma=1 beta=0, no relu
  matvec_wmma_part<<<dim3(2048 / TILE_COLS, KSPLIT), 32, 0, stream>>>(
      y2, W3, part, 8192 / KSPLIT, 2048);
  finalize_bn<<<(2048 + 255) / 256, 256, 0, stream>>>(
      part, 2048, KSPLIT, b3, nullptr, nullptr, m3, v3, mk, 0);

  // Output must start at 0 (harness poisons it); then running max over samples.
  zero_f32<<<(NCLASS + 255) / 256, 256, 0, stream>>>(z, NCLASS);
  gumbel_st_max<<<1024, GT, 0, stream>>>(mk, u, z);
}